// Lacunarity_31885837205967
// MI455X (gfx1250) — compile-verified
//
#include <hip/hip_runtime.h>

// CDNA5 / gfx1250 — wave32, WMMA (not MFMA).
typedef __attribute__((ext_vector_type(2))) float v2f;
typedef __attribute__((ext_vector_type(4))) float v4f;
typedef __attribute__((ext_vector_type(8))) float v8f;

#define M_DIM 4096            // reduced axis
#define N_DIM 1024            // prob axis (p = n+1)
#define ROWS_PER_BLOCK 16
#define THREADS 256           // 8 wave32 waves per block

__global__ __launch_bounds__(THREADS)
void lacunarity_kernel(const float* __restrict__ in, float* __restrict__ out) {
    // Padded stride 18: even (8B-aligned b64 reads) and conflict-friendly.
    __shared__ float part[ROWS_PER_BLOCK][18];

    const int tid = threadIdx.x;
    const int row_in_tile = tid >> 4;     // 0..15
    const int sub = tid & 15;             // 16 threads cooperate per row
    const long long row = (long long)blockIdx.x * ROWS_PER_BLOCK + row_in_tile;
    const float* rp = in + row * (long long)M_DIM + sub * 4;

    // Phase 1: stream 4096 floats/row. Per iteration a wave issues b128 loads
    // covering two fully-used 256B contiguous segments (16 lanes x 16B / row).
    v4f a0 = {0.f, 0.f, 0.f, 0.f};
    v4f a1 = {0.f, 0.f, 0.f, 0.f};
    v4f a2 = {0.f, 0.f, 0.f, 0.f};
    v4f a3 = {0.f, 0.f, 0.f, 0.f};
#pragma unroll 4
    for (int i = 0; i < 64; i += 4) {
        v4f x0 = *(const v4f*)(rp + (i + 0) * 64);
        v4f x1 = *(const v4f*)(rp + (i + 1) * 64);
        v4f x2 = *(const v4f*)(rp + (i + 2) * 64);
        v4f x3 = *(const v4f*)(rp + (i + 3) * 64);
        a0 += x0; a1 += x1; a2 += x2; a3 += x3;
    }
    v4f a = (a0 + a1) + (a2 + a3);
    part[row_in_tile][sub] = (a.x + a.y) + (a.z + a.w);
    __syncthreads();

    // Phase 2: reduce the 16x16 partial matrix with WMMA:
    //   D = A(16x4) * ones(4x16) + C, accumulated over K=16 in 4 chunks.
    // Every wave executes it (EXEC must be all-1s for WMMA); results identical.
    // A-fragment (16x4 f32): lanes 0-15 -> K = k0,k0+1 ; lanes 16-31 -> K = k0+2,k0+3.
    const int lane = tid & 31;
    const int r16 = lane & 15;
    const int half = lane >> 4;
    const v2f ones2 = {1.0f, 1.0f};       // B = all ones (layout-independent)
    v8f d = {0.f, 0.f, 0.f, 0.f, 0.f, 0.f, 0.f, 0.f};
#pragma unroll
    for (int k0 = 0; k0 < 16; k0 += 4) {
        v2f afrag = *(const v2f*)&part[r16][k0 + 2 * half];
        d = __builtin_amdgcn_wmma_f32_16x16x4_f32(
                /*neg_a=*/false, afrag, /*neg_b=*/false, ones2,
                /*c_mod=*/(short)0, d, /*reuse_a=*/false, /*reuse_b=*/false);
    }

    // D column 0: lane 0 VGPR j = sum(row j), lane 16 VGPR j = sum(row j+8).
    if (tid == 0 || tid == 16) {
        const int mbase = (tid == 16) ? 8 : 0;
        const long long row0 = (long long)blockIdx.x * ROWS_PER_BLOCK;
#pragma unroll
        for (int j = 0; j < 8; ++j) {
            long long g = row0 + mbase + j;
            float s = d[j];
            float p = (float)((int)(g & (N_DIM - 1)) + 1);   // prob = n+1
            float mu1 = p * s;
            float mu1sq = mu1 * mu1;
            float mu2 = p * p * s;
            out[g] = (mu2 - mu1sq) / mu1sq;                  // == (1-s)/s
        }
    }
}

extern "C" void kernel_launch(void* const* d_in, const int* in_sizes, int n_in,
                              void* d_out, int out_size, void* d_ws, size_t ws_size,
                              hipStream_t stream) {
    (void)n_in; (void)out_size; (void)d_ws; (void)ws_size;
    const float* in = (const float*)d_in[0];
    float* out = (float*)d_out;
    const long long rows = (long long)in_sizes[0] / M_DIM;   // 32768
    const int grid = (int)(rows / ROWS_PER_BLOCK);           // 2048
    lacunarity_kernel<<<grid, THREADS, 0, stream>>>(in, out);
}